// dfsphModule_42382737277398
// MI455X (gfx1250) — compile-verified
//
#include <hip/hip_runtime.h>
#include <hip/hip_bf16.h>

#define TILE 256
#define C_WENDLAND 2.2281692032865347f /* 7/pi */

#define AS1 __attribute__((address_space(1)))
#define AS3 __attribute__((address_space(3)))
typedef int v2i __attribute__((ext_vector_type(2)));

// ---- CDNA5 async global->LDS copies (ASYNCcnt), non-temporal (TH=NT) --------
__device__ __forceinline__ void async_copy_b32(const void* g, void* l) {
#if __has_builtin(__builtin_amdgcn_global_load_async_to_lds_b32)
    __builtin_amdgcn_global_load_async_to_lds_b32(
        (AS1 int*)g, (AS3 int*)l, 0, /*cpol: TH_NT*/ 1);
#else
    asm volatile("global_load_async_to_lds_b32 %0, %1, off"
                 :: "v"((unsigned)(unsigned long long)l), "v"(g) : "memory");
#endif
}

__device__ __forceinline__ void async_copy_b64(const void* g, void* l) {
#if __has_builtin(__builtin_amdgcn_global_load_async_to_lds_b64)
    __builtin_amdgcn_global_load_async_to_lds_b64(
        (AS1 v2i*)g, (AS3 v2i*)l, 0, /*cpol: TH_NT*/ 1);
#else
    asm volatile("global_load_async_to_lds_b64 %0, %1, off"
                 :: "v"((unsigned)(unsigned long long)l), "v"(g) : "memory");
#endif
}

template <int CNT>
__device__ __forceinline__ void wait_asynccnt() {
#if __has_builtin(__builtin_amdgcn_s_wait_asynccnt)
    __builtin_amdgcn_s_wait_asynccnt(CNT);
#else
    asm volatile("s_wait_asynccnt %0" :: "i"(CNT) : "memory");
#endif
}

// Hardware f32 atomic add (global_atomic_add_f32), no CAS expansion.
__device__ __forceinline__ void atomAdd(float* p, float v) {
    unsafeAtomicAdd(p, v);
}

// ---- Pass 2 (defined first so disasm snippet shows it): edge scatter --------
__global__ __launch_bounds__(TILE) void dfsph_scatter(
    const float*  __restrict__ rad,   const float2* __restrict__ dist,
    const int*    __restrict__ nbrI,  const int*    __restrict__ nbrJ,
    const float4* __restrict__ packed, const float* __restrict__ pArr,
    float* __restrict__ k1x, float* __restrict__ k1y, float* __restrict__ k2,
    float* __restrict__ axA, float* __restrict__ ayA,
    float* __restrict__ s2A, float* __restrict__ kkA,
    const float* __restrict__ sup, int E, int nTiles, int tilesPerBlock)
{
    __shared__ float  sRad[2][TILE];
    __shared__ float2 sDist[2][TILE];
    __shared__ int    sI[2][TILE];
    __shared__ int    sJ[2][TILE];

    const int tid = threadIdx.x;
    int t0 = (int)blockIdx.x * tilesPerBlock;
    int t1 = min(t0 + tilesPerBlock, nTiles);
    if (t0 >= t1) return;

    float h    = *sup;
    float coef = 20.0f * C_WENDLAND / (h * h * h);

    auto issue = [&](int t, int buf) {
        int e = t * TILE + tid;
        if (e < E) {
            async_copy_b32(rad  + e, &sRad[buf][tid]);
            async_copy_b64(dist + e, &sDist[buf][tid]);
            async_copy_b32(nbrI + e, &sI[buf][tid]);
            async_copy_b32(nbrJ + e, &sJ[buf][tid]);
        }
    };

    issue(t0, 0);
    for (int t = t0; t < t1; ++t) {
        int  buf  = (t - t0) & 1;
        bool more = (t + 1 < t1);
        if (more) {
            issue(t + 1, buf ^ 1);     // 8 ops in flight (2 tiles)
            wait_asynccnt<4>();        // drain oldest tile (in-order completion)
        } else {
            wait_asynccnt<0>();
        }
        int e = t * TILE + tid;
        if (e < E) {
            float  q   = sRad[buf][tid];
            float2 dxy = sDist[buf][tid];
            int    i   = sI[buf][tid];
            int    j   = sJ[buf][tid];

            float u   = 1.0f - q;
            float mag = coef * q * u * u * u;
            float gx  = -dxy.x * mag;
            float gy  = -dxy.y * mag;
            float g2  = gx * gx + gy * gy;

            float4 pj0 = packed[2 * j];       // {aa, A2, p, m*rho0}
            float4 pj1 = packed[2 * j + 1];   // {vx, vy, ax, ay}
            float  pi  = pArr[i];
            float  aaj = pj0.x;

            atomAdd(&k1x[i], aaj * gx);
            atomAdd(&k1y[i], aaj * gy);
            atomAdd(&k2[i],  pj0.y * g2);
            float fps = -pj0.w * (pi + pj0.z);
            atomAdd(&axA[i], fps * gx);
            atomAdd(&ayA[i], fps * gy);
            atomAdd(&s2A[i], aaj * (pj1.x * gx + pj1.y * gy));
            atomAdd(&kkA[i], aaj * (pj1.z * gx + pj1.w * gy));
        }
    }
}

// ---- Pass 1: per-particle precompute + accumulator zeroing ------------------
__global__ __launch_bounds__(256) void dfsph_prep(
    const float* __restrict__ area, const float* __restrict__ aarea,
    const float* __restrict__ rho0, const float* __restrict__ dens,
    const float* __restrict__ pres, const float* __restrict__ vel,
    const float* __restrict__ pacc,
    float4* __restrict__ packed,  // [2N]: {aa, A2, p, m*rho0}, {vx, vy, ax, ay}
    float*  __restrict__ pArr,    // [N]
    float*  __restrict__ accum,   // [7N]: k1x,k1y,k2,accx,accy,s2,kk2
    int N)
{
    int n = blockIdx.x * blockDim.x + threadIdx.x;
    if (n >= N) return;
    float a  = area[n];
    float aa = aarea[n];
    float r0 = rho0[n];
    float mAr = a * r0;
    float A2  = (aa * aa) / mAr;            // actualArea^2 / (area*restDensity)
    float dr  = dens[n] * r0;
    float p   = pres[n] / (dr * dr);        // pressure2 / (density*restDensity)^2
    packed[2 * n]     = make_float4(aa, A2, p, mAr);
    packed[2 * n + 1] = make_float4(vel[2 * n], vel[2 * n + 1],
                                    pacc[2 * n], pacc[2 * n + 1]);
    pArr[n] = p;
#pragma unroll
    for (int c = 0; c < 7; ++c) accum[(size_t)c * N + n] = 0.0f;
}

// ---- Pass 3: per-particle finalize -----------------------------------------
__global__ __launch_bounds__(256) void dfsph_finalize(
    const float* __restrict__ area, const float* __restrict__ aarea,
    const float* __restrict__ rho0, const float* __restrict__ vel,
    const float* __restrict__ pacc,
    const float* __restrict__ k1x, const float* __restrict__ k1y,
    const float* __restrict__ k2,  const float* __restrict__ axA,
    const float* __restrict__ ayA, const float* __restrict__ s2A,
    const float* __restrict__ kkA,
    const float* __restrict__ dtp, float* __restrict__ out, int N)
{
    int n = blockIdx.x * blockDim.x + threadIdx.x;
    if (n >= N) return;
    float dt   = *dtp;
    float kx   = k1x[n], ky = k1y[n];
    float fac  = -dt * dt * aarea[n];
    float mass = area[n] * rho0[n];
    float alpha = (fac / mass) * (kx * kx + ky * ky) + fac * k2[n];
    float vx = vel[2 * n], vy = vel[2 * n + 1];
    float axn = pacc[2 * n], ayn = pacc[2 * n + 1];
    float src = -dt * (vx * kx + vy * ky - s2A[n]);
    float ks  = dt * dt * (axn * kx + ayn * ky - kkA[n]);
    out[5 * n + 0] = alpha;
    out[5 * n + 1] = axA[n];
    out[5 * n + 2] = ayA[n];
    out[5 * n + 3] = src;
    out[5 * n + 4] = ks;
}

extern "C" void kernel_launch(void* const* d_in, const int* in_sizes, int n_in,
                              void* d_out, int out_size, void* d_ws, size_t ws_size,
                              hipStream_t stream) {
    const float*  area  = (const float*)d_in[0];
    const float*  aarea = (const float*)d_in[1];
    const float*  rho0  = (const float*)d_in[2];
    const float*  dens  = (const float*)d_in[3];
    const float*  pres  = (const float*)d_in[4];
    const float*  vel   = (const float*)d_in[5];
    const float*  pacc  = (const float*)d_in[6];
    const float*  rad   = (const float*)d_in[7];
    const float2* dist  = (const float2*)d_in[8];
    const int*    nbr   = (const int*)d_in[9];
    const float*  sup   = (const float*)d_in[10];
    const float*  dtp   = (const float*)d_in[11];

    int N = in_sizes[0];
    int E = in_sizes[7];
    const int* nbrI = nbr;
    const int* nbrJ = nbr + E;

    float*  ws     = (float*)d_ws;
    float4* packed = (float4*)ws;                 // 8N floats
    float*  pArr   = ws + (size_t)8 * N;          // N
    float*  accum  = ws + (size_t)9 * N;          // 7N
    float*  k1x = accum + (size_t)0 * N;
    float*  k1y = accum + (size_t)1 * N;
    float*  k2  = accum + (size_t)2 * N;
    float*  axA = accum + (size_t)3 * N;
    float*  ayA = accum + (size_t)4 * N;
    float*  s2A = accum + (size_t)5 * N;
    float*  kkA = accum + (size_t)6 * N;

    dfsph_prep<<<(N + 255) / 256, 256, 0, stream>>>(
        area, aarea, rho0, dens, pres, vel, pacc, packed, pArr, accum, N);

    int nTiles        = (E + TILE - 1) / TILE;
    int tilesPerBlock = 16;                       // contiguous tiles per block
    int grid          = (nTiles + tilesPerBlock - 1) / tilesPerBlock;
    dfsph_scatter<<<grid, TILE, 0, stream>>>(
        rad, dist, nbrI, nbrJ, packed, pArr,
        k1x, k1y, k2, axA, ayA, s2A, kkA, sup, E, nTiles, tilesPerBlock);

    dfsph_finalize<<<(N + 255) / 256, 256, 0, stream>>>(
        area, aarea, rho0, vel, pacc, k1x, k1y, k2, axA, ayA, s2A, kkA,
        dtp, (float*)d_out, N);
}